// MultiScaleDeformableAttention_49795850830056
// MI455X (gfx1250) — compile-verified
//
#include <hip/hip_runtime.h>
#include <hip/hip_bf16.h>
#include <math.h>

// Problem constants (from the reference file)
#define BS_    4
#define NQ_    4000
#define EMBED_ 256
#define NH_    8
#define NL_    4
#define NP_    4
#define HD_    32
#define NV_    13294   // 100*100 + 50*50 + 25*25 + 13*13

typedef __attribute__((ext_vector_type(2))) float v2f;
typedef __attribute__((ext_vector_type(8))) float v8f;
typedef unsigned int v4u __attribute__((ext_vector_type(4)));
typedef int v4i __attribute__((ext_vector_type(4)));
typedef int v8i __attribute__((ext_vector_type(8)));

// ---------------------------------------------------------------------------
// FP32 WMMA GEMM: C[M,N] = A[M,K] @ B[K,N] + bias[N]
//
// Block = 8 waves = 256 threads, computes a 128(M) x 64(N) tile.
// The K x 64 slice of B (the weight matrix) is staged into LDS once per
// block by the Tensor Data Mover (tensor_load_to_lds, TENSORcnt), then each
// wave computes a 16x64 strip with V_WMMA_F32_16X16X4_F32, assembling B
// fragments from LDS and streaming A from global with an incremented pointer.
//
// Bt is DYNAMIC LDS (extern __shared__) + its address is escaped into an
// inline asm with a memory clobber: the TDM builtin only consumes descriptor
// values, so with a static internal LDS array the compiler proved "no stores
// exist" and folded every Bt read to undef (observed in round 2/3 asm). The
// external dynamic-LDS symbol + escaped pointer make that fold impossible.
//
// Fragment layouts (ISA 7.12.2):
//   A 16x4 : lanes 0-15 -> M=lane, VGPR0=K0, VGPR1=K1; lanes 16-31 -> K2/K3.
//   B 4x16 : lane -> N, same half-wave K split.
//   C/D    : VGPR r -> row r (lanes 0-15) / r+8 (lanes 16-31), N = lane&15.
// ---------------------------------------------------------------------------
__global__ void __launch_bounds__(256) gemm_f32_wmma(
    const float* __restrict__ A, const float* __restrict__ B,
    const float* __restrict__ bias, float* __restrict__ C,
    int M, int N, int K)
{
    extern __shared__ float Bt[];          // K x 64 tile of B (64 KB dynamic)

    const int ngroups = N >> 6;            // 64-wide column groups
    const int mb = blockIdx.x / ngroups;   // 128-row block index
    const int ng = blockIdx.x - mb * ngroups;
    const int n0 = ng << 6;

    const int wv   = threadIdx.x >> 5;     // wave in block (0..7)
    const int lane = threadIdx.x & 31;
    const int half = lane >> 4;            // 0: K={0,1}, 1: K={2,3}
    const int l15  = lane & 15;

    // ---- stage B[K x 64] tile into LDS via the Tensor Data Mover ----------
    if (wv == 0) {
        const uint64_t ga = (uint64_t)(uintptr_t)(B + n0);
        v4u g0;
        g0[0] = 1u;                                        // count=1 (valid)
        g0[1] = 0u;                                        // lds_addr = 0 (Bt)
        g0[2] = (unsigned)(ga & 0xFFFFFFFFu);              // global_addr[31:0]
        g0[3] = (unsigned)((ga >> 32) & 0x01FFFFFFu)       // global_addr[56:32]
              | (2u << 30);                                // type = 2 (image)

        const unsigned d0 = (unsigned)(N - n0);            // tensor_dim0
        const unsigned d1 = (unsigned)K;                   // tensor_dim1
        v8i g1;
        g1[0] = (int)(2u << 16);                           // data_size = 4B
        g1[1] = (int)((d0 & 0xFFFFu) << 16);               // dim0[15:0]
        g1[2] = (int)(((d0 >> 16) & 0xFFFFu)               // dim0[31:16]
              | ((d1 & 0xFFFFu) << 16));                   // dim1[15:0]
        g1[3] = (int)(((d1 >> 16) & 0xFFFFu)               // dim1[31:16]
              | (64u << 16));                              // tile_dim0 = 64
        g1[4] = (int)((unsigned)K & 0xFFFFu);              // tile_dim1=K, tile_dim2=0
        g1[5] = N;                                         // dim0_stride[31:0]
        g1[6] = 0;                                         // stride hi / dim1_stride lo
        g1[7] = 0;
        v4i gz = {0, 0, 0, 0};
#if __clang_major__ >= 23
        v8i gz8 = {0, 0, 0, 0, 0, 0, 0, 0};
        __builtin_amdgcn_tensor_load_to_lds(g0, g1, gz, gz, gz8, 0);
#else
        __builtin_amdgcn_tensor_load_to_lds(g0, g1, gz, gz, 0);
#endif
        __builtin_amdgcn_s_wait_tensorcnt(0);
    }
    // Escape Bt's address into a may-write asm so the compiler cannot prove
    // "no stores to Bt exist" (the TDM writes it behind the compiler's back).
    {
        void* btesc = (void*)Bt;
        asm volatile("" : "+v"(btesc) :: "memory");
    }
    __syncthreads();

    // ---- per-wave 16x64 strip --------------------------------------------
    const int m0 = mb * 128 + wv * 16;
    int arow = m0 + l15;                   // clamp tail rows; stores masked
    if (arow >= M) arow = M - 1;
    const float* __restrict__ ap = A + (size_t)arow * K + (half << 1);

    int kb = (half << 1) * 64 + l15;       // LDS fragment base index

    v8f acc0 = {}; v8f acc1 = {}; v8f acc2 = {}; v8f acc3 = {};

    #pragma unroll 4
    for (int kk = 0; kk < K; kk += 4) {
        v2f a; a.x = ap[0]; a.y = ap[1]; ap += 4;

        v2f b0; b0.x = Bt[kb];      b0.y = Bt[kb + 64];
        v2f b1; b1.x = Bt[kb + 16]; b1.y = Bt[kb + 80];
        v2f b2; b2.x = Bt[kb + 32]; b2.y = Bt[kb + 96];
        v2f b3; b3.x = Bt[kb + 48]; b3.y = Bt[kb + 112];
        kb += 256;

        acc0 = __builtin_amdgcn_wmma_f32_16x16x4_f32(false, a, false, b0, (short)0, acc0, false, false);
        acc1 = __builtin_amdgcn_wmma_f32_16x16x4_f32(false, a, false, b1, (short)0, acc1, false, false);
        acc2 = __builtin_amdgcn_wmma_f32_16x16x4_f32(false, a, false, b2, (short)0, acc2, false, false);
        acc3 = __builtin_amdgcn_wmma_f32_16x16x4_f32(false, a, false, b3, (short)0, acc3, false, false);
    }

    const float bb0 = bias[n0 + l15];
    const float bb1 = bias[n0 + l15 + 16];
    const float bb2 = bias[n0 + l15 + 32];
    const float bb3 = bias[n0 + l15 + 48];

    #pragma unroll
    for (int r = 0; r < 8; ++r) {
        const int row = m0 + r + (half << 3);
        if (row < M) {
            float* __restrict__ Cr = C + (size_t)row * N + n0 + l15;
            Cr[0]  = acc0[r] + bb0;
            Cr[16] = acc1[r] + bb1;
            Cr[32] = acc2[r] + bb2;
            Cr[48] = acc3[r] + bb3;
        }
    }
}

// ---------------------------------------------------------------------------
// Sampling kernel: one wave per (b, q, h).
// Lanes 0-15 own one (level, point): compute pixel coords + softmax weight.
// Then loop over the 16 points; lane = channel (HD=32), so each bilinear
// corner gather is one coalesced 128B read of V[b, pos, h*32 + lane].
// ---------------------------------------------------------------------------
__global__ void __launch_bounds__(256) msda_sample(
    const float* __restrict__ V,     // [BS, NV, 256]   (channel = h*32+hd)
    const float* __restrict__ OFF,   // [BS, NQ, NH, 16, 2]
    const float* __restrict__ ATT,   // [BS, NQ, NH, 16] (logits)
    const float* __restrict__ REF,   // [BS, NQ, NL, 2]
    float* __restrict__ OUT)         // [BS, NQ, NH, HD] = [16000, 256]
{
    const int lane = threadIdx.x & 31;
    const int wave = (int)((blockIdx.x * blockDim.x + threadIdx.x) >> 5);
    if (wave >= BS_ * NQ_ * NH_) return;

    const int h = wave & 7;
    const int q = (wave >> 3) % NQ_;
    const int b = wave / (NQ_ * NH_);

    const int l15 = lane & 15;
    const int lvl = l15 >> 2;

    const size_t qb = (size_t)b * NQ_ + q;

    // per-(l,p) params, replicated in both half-waves
    const float logit = ATT[(qb * NH_ + h) * 16 + l15];
    const float offx  = OFF[((qb * NH_ + h) * 16 + l15) * 2 + 0];
    const float offy  = OFF[((qb * NH_ + h) * 16 + l15) * 2 + 1];
    const float refx  = REF[(qb * NL_ + lvl) * 2 + 0];
    const float refy  = REF[(qb * NL_ + lvl) * 2 + 1];

    const float Wl = (lvl == 0) ? 100.f : (lvl == 1) ? 50.f : (lvl == 2) ? 25.f : 13.f;
    // x = ref_x*W + off_x - 0.5 ;  y = ref_y*H + off_y - 0.5  (H==W per level)
    const float x = refx * Wl + offx - 0.5f;
    const float y = refy * Wl + offy - 0.5f;

    // softmax over the 16 logits (both halves hold identical copies, so
    // xor reductions over masks 8..1 reduce each 16-lane group correctly)
    float mx = logit;
    #pragma unroll
    for (int s = 8; s >= 1; s >>= 1) mx = fmaxf(mx, __shfl_xor(mx, s, 32));
    const float e = __expf(logit - mx);
    float ssum = e;
    #pragma unroll
    for (int s = 8; s >= 1; s >>= 1) ssum += __shfl_xor(ssum, s, 32);
    const float w = e / ssum;

    float acc = 0.0f;

    #pragma unroll
    for (int p = 0; p < 16; ++p) {
        const float xp = __shfl(x, p, 32);
        const float yp = __shfl(y, p, 32);
        const float wp = __shfl(w, p, 32);

        const int plvl  = p >> 2;
        const int Wd    = (plvl == 0) ? 100 : (plvl == 1) ? 50 : (plvl == 2) ? 25 : 13;
        const int start = (plvl == 0) ? 0 : (plvl == 1) ? 10000 : (plvl == 2) ? 12500 : 13125;

        const float fx0 = floorf(xp), fy0 = floorf(yp);
        const int ix = (int)fx0, iy = (int)fy0;
        const float tx = xp - fx0, ty = yp - fy0;

        const float* __restrict__ Vb =
            V + ((size_t)b * NV_ + start) * EMBED_ + h * HD_ + lane;

        #pragma unroll
        for (int c = 0; c < 4; ++c) {
            const int dx = c & 1, dy = c >> 1;
            const int cx = ix + dx, cy = iy + dy;
            const float wc = (dx ? tx : 1.0f - tx) * (dy ? ty : 1.0f - ty);
            const bool valid = (cx >= 0) & (cx < Wd) & (cy >= 0) & (cy < Wd);
            const int ccx = cx < 0 ? 0 : (cx > Wd - 1 ? Wd - 1 : cx);
            const int ccy = cy < 0 ? 0 : (cy > Wd - 1 ? Wd - 1 : cy);
            const float val = Vb[(size_t)(ccy * Wd + ccx) * EMBED_];
            acc += wp * wc * (valid ? val : 0.0f);
        }
    }

    OUT[(qb * NH_ + h) * HD_ + lane] = acc;
}

// ---------------------------------------------------------------------------
extern "C" void kernel_launch(void* const* d_in, const int* in_sizes, int n_in,
                              void* d_out, int out_size, void* d_ws, size_t ws_size,
                              hipStream_t stream)
{
    const float* query  = (const float*)d_in[0];
    const float* value  = (const float*)d_in[1];
    const float* refp   = (const float*)d_in[2];
    // d_in[3] = spatial_shapes (int32) -- compile-time constants here
    const float* W_off  = (const float*)d_in[4];
    const float* b_off  = (const float*)d_in[5];
    const float* W_attn = (const float*)d_in[6];
    const float* b_attn = (const float*)d_in[7];
    const float* W_val  = (const float*)d_in[8];
    const float* b_val  = (const float*)d_in[9];
    const float* W_out  = (const float*)d_in[10];
    const float* b_out  = (const float*)d_in[11];
    float* out = (float*)d_out;

    // workspace layout (all fp32)
    float* V   = (float*)d_ws;                               // BS*NV*256
    float* OFF = V   + (size_t)BS_ * NV_ * EMBED_;           // BS*NQ*256
    float* ATT = OFF + (size_t)BS_ * NQ_ * EMBED_;           // BS*NQ*128
    float* TMP = ATT + (size_t)BS_ * NQ_ * 128;              // BS*NQ*256

    auto gemm = [&](const float* A, const float* B, const float* bias,
                    float* C, int M, int N, int K) {
        const int mblocks = (M + 127) >> 7;                  // 128 rows / block
        const int blocks  = mblocks * (N >> 6);
        const size_t ldsB = (size_t)K * 64 * sizeof(float);  // dynamic LDS tile
        gemm_f32_wmma<<<blocks, 256, ldsB, stream>>>(A, B, bias, C, M, N, K);
    };

    gemm(value, W_val,  b_val,  V,   BS_ * NV_, 256, 256);   // value projection
    gemm(query, W_off,  b_off,  OFF, BS_ * NQ_, 256, 256);   // sampling offsets
    gemm(query, W_attn, b_attn, ATT, BS_ * NQ_, 128, 256);   // attention logits

    const int swaves = BS_ * NQ_ * NH_;                      // 128000 waves
    msda_sample<<<(swaves + 7) / 8, 256, 0, stream>>>(V, OFF, ATT, refp, TMP);

    gemm(TMP, W_out, b_out, out, BS_ * NQ_, 256, 256);       // output projection
}